// LLMFMLPrec_33406255628492
// MI455X (gfx1250) — compile-verified
//
#include <hip/hip_runtime.h>
#include <math.h>

// ---------------------------------------------------------------------------
// LLMFMLPrec on MI455X (gfx1250, wave32).
// All matrix ops (GEMMs, attention scores, attention context) run on
// V_WMMA_F32_16X16X4_F32: full fp32 precision matching the reference, on the
// CDNA5 matrix pipe.
// GNN is algebraically collapsed (adj @ Y == m_i * masked_mean(Y)); the S=200
// encoder is the causal prefix of the S=201 encoder, so the transformer runs
// only once.
// ---------------------------------------------------------------------------

typedef float v2f __attribute__((ext_vector_type(2)));
typedef float v8f __attribute__((ext_vector_type(8)));

#define B_SZ   256
#define L1_SZ  201
#define LEN    200
#define DLLM   768
#define HDIM   64
#define NHEAD  2
#define DHEAD  32
#define FFN_D  256
#define ROWS   (B_SZ * L1_SZ)   // 51456 = 16 * 3216

// attention tiling
#define QTILES 13     // ceil(201/16)
#define SPAD   208    // keys padded to 13*16
#define KSTR   34     // Q/K LDS row stride (even -> b64-aligned pairs, conflict-free)
#define PSTR   210    // P scratch / V^T row stride (even, conflict-free)

__device__ __forceinline__ v8f wmma4(v2f a, v2f b, v8f c) {
  // D(16x16,f32) = A(16x4,f32) * B(4x16,f32) + C
  return __builtin_amdgcn_wmma_f32_16x16x4_f32(false, a, false, b, (short)0, c,
                                               false, false);
}

__device__ __forceinline__ float wave_sum(float v) {
#pragma unroll
  for (int off = 16; off > 0; off >>= 1) v += __shfl_xor(v, off, 32);
  return v;
}
__device__ __forceinline__ float wave_max(float v) {
#pragma unroll
  for (int off = 16; off > 0; off >>= 1) v = fmaxf(v, __shfl_xor(v, off, 32));
  return v;
}

__device__ __forceinline__ float apply_act(float v, int act) {
  if (act == 1) return 0.5f * v * (1.0f + erff(v * 0.70710678118654752f)); // exact gelu
  if (act == 2) return fmaxf(v, 0.0f);                                    // relu
  return v;
}

// ---------------------------------------------------------------------------
// Generic WMMA fp32 GEMM: C = act(A(MxK) @ W(KxN) + bias), row-major.
// Requires M%16==0, N%64==0, K%4==0 (true for every GEMM in this model).
// Fragment layouts per CDNA5 ISA 7.12.2:
//   A 16x4 : lane = m + 16*kh ; v = { A[m][k+2kh], A[m][k+2kh+1] }
//   B 4x16 : lane = n + 16*kh ; v = { B[k+2kh][n], B[k+2kh+1][n] }
//   C 16x16: VGPR r, lanes 0-15 -> row r, lanes 16-31 -> row r+8
// ---------------------------------------------------------------------------
__device__ __forceinline__ void store_tile(float* __restrict__ C,
                                           const float* __restrict__ bias,
                                           int act, v8f acc, int mt, int ncol0,
                                           int N, int lane) {
  int col = ncol0 + (lane & 15);
  int rbase = mt * 16 + ((lane >> 4) << 3);
  float bb = bias ? bias[col] : 0.0f;
#pragma unroll
  for (int r = 0; r < 8; ++r) {
    C[(size_t)(rbase + r) * N + col] = apply_act(acc[r] + bb, act);
  }
}

__global__ __launch_bounds__(256) void wmma_gemm_kernel(
    const float* __restrict__ A, const float* __restrict__ W,
    const float* __restrict__ bias, float* __restrict__ C, int M, int N, int K,
    int act) {
  int lane = threadIdx.x & 31;
  int wave = (int)((blockIdx.x * blockDim.x + threadIdx.x) >> 5);
  int nwaves = (int)((gridDim.x * blockDim.x) >> 5);
  int mrow = lane & 15;
  int kh = lane >> 4;  // which K-half this lane holds
  int mtiles = M >> 4;

  for (int mt = wave; mt < mtiles; mt += nwaves) {
    const float* Ap = A + (size_t)(mt * 16 + mrow) * K + 2 * kh;
    for (int n0 = 0; n0 < N; n0 += 64) {
      v8f acc0 = {}, acc1 = {}, acc2 = {}, acc3 = {};
      int col = n0 + (lane & 15);
      for (int k = 0; k < K; k += 4) {
        v2f a = *(const v2f*)(Ap + k);  // 8B-aligned contiguous pair
        const float* W0 = W + (size_t)(k + 2 * kh) * N + col;
        const float* W1 = W0 + N;
        v2f b0 = {W0[0], W1[0]};
        v2f b1 = {W0[16], W1[16]};
        v2f b2 = {W0[32], W1[32]};
        v2f b3 = {W0[48], W1[48]};
        acc0 = wmma4(a, b0, acc0);
        acc1 = wmma4(a, b1, acc1);
        acc2 = wmma4(a, b2, acc2);
        acc3 = wmma4(a, b3, acc3);
      }
      store_tile(C, bias, act, acc0, mt, n0 + 0, N, lane);
      store_tile(C, bias, act, acc1, mt, n0 + 16, N, lane);
      store_tile(C, bias, act, acc2, mt, n0 + 32, N, lane);
      store_tile(C, bias, act, acc3, mt, n0 + 48, N, lane);
    }
  }
}

// ---------------------------------------------------------------------------
// WMMA attention: one block (4 waves) per (batch, head).
// Q,K row-major and V transposed are staged in LDS with even, conflict-free
// strides so every fragment fetch is a single aligned ds_load_b64.
// Per 16-query tile: scores = Q@K^T via WMMA -> per-wave LDS scratch,
// softmax in LDS with exact reference semantics (scale + -1e9 bias over all
// 201 keys), ctx = P@V via WMMA, scaled by 1/rowsum at store.
// ---------------------------------------------------------------------------
__global__ __launch_bounds__(128) void attention_wmma_kernel(
    const float* __restrict__ Qg, const float* __restrict__ Kg,
    const float* __restrict__ Vg, const int* __restrict__ maskg,
    float* __restrict__ Cg) {
  __shared__ float Qs[SPAD * KSTR];     // 28.3 KB
  __shared__ float Ksh[SPAD * KSTR];    // 28.3 KB
  __shared__ float Vt[DHEAD * PSTR];    // 26.9 KB (V transposed: [dim][key])
  __shared__ float Ps[4][16 * PSTR];    // 53.8 KB per-wave score scratch
  __shared__ float Sv[4][16];           // per-row 1/sum
  __shared__ float Ms[SPAD];

  int b = blockIdx.x >> 1;
  int h = blockIdx.x & 1;
  int tid = threadIdx.x;

  // zero only the padded regions (rows/cols >= 201) -- race-free vs staging
  for (int idx = tid; idx < (SPAD - 201) * KSTR; idx += 128) {
    Qs[201 * KSTR + idx] = 0.0f;
    Ksh[201 * KSTR + idx] = 0.0f;
  }
  for (int idx = tid; idx < DHEAD * 8; idx += 128) {
    int d = idx >> 3;
    Vt[d * PSTR + 201 + (idx & 7)] = 0.0f;  // key cols 201..208
  }
  for (int idx = tid; idx < L1_SZ * DHEAD; idx += 128) {
    int s = idx >> 5, d = idx & 31;
    size_t g = (size_t)(b * L1_SZ + s) * HDIM + h * DHEAD + d;
    Qs[s * KSTR + d] = Qg[g];
    Ksh[s * KSTR + d] = Kg[g];
    Vt[d * PSTR + s] = Vg[g];
  }
  for (int j = tid; j < SPAD; j += 128)
    Ms[j] = (j < L1_SZ) ? (float)maskg[b * L1_SZ + j] : 0.0f;
  __syncthreads();

  int w = tid >> 5, lane = tid & 31;
  int nn = lane & 15, kh = lane >> 4;
  const float scale = 0.17677669529663687f;  // 1/sqrt(32)
  float* Pw = Ps[w];

  for (int qt = w; qt < QTILES; qt += 4) {
    int q0 = qt * 16;
    // ---- scores: S(16x16 tiles) = Q(16x32) @ K^T
    const float* qa = Qs + (q0 + nn) * KSTR + 2 * kh;  // A: row q0+m
    for (int kt = 0; kt < QTILES; ++kt) {
      int k0 = kt * 16;
      const float* kb = Ksh + (k0 + nn) * KSTR + 2 * kh;  // B col n = K row k0+n
      v8f acc = {};
#pragma unroll
      for (int kk = 0; kk < DHEAD; kk += 4) {
        v2f a = *(const v2f*)(qa + kk);
        v2f bf = *(const v2f*)(kb + kk);
        acc = wmma4(a, bf, acc);
      }
#pragma unroll
      for (int r = 0; r < 8; ++r)
        Pw[(r + 8 * kh) * PSTR + k0 + nn] = acc[r];
    }
    // ---- softmax (raw -> exp, pads -> 0), exact reference math
    for (int r = 0; r < 16; ++r) {
      int q = q0 + r;
      float* row = Pw + r * PSTR;
      float mx = -3.0e38f;
      for (int j = lane; j < L1_SZ; j += 32) {
        float sc = row[j] * scale + ((j <= q && Ms[j] != 0.0f) ? 0.0f : -1.0e9f);
        row[j] = sc;
        mx = fmaxf(mx, sc);
      }
      mx = wave_max(mx);
      float sum = 0.0f;
      for (int j = lane; j < SPAD; j += 32) {
        float e = (j < L1_SZ) ? expf(row[j] - mx) : 0.0f;
        row[j] = e;
        sum += e;
      }
      sum = wave_sum(sum);
      if (lane == 0) Sv[w][r] = 1.0f / sum;
    }
    // ---- ctx: C(16x32) = P(16x208) @ V(208x32)
    v8f c0 = {}, c1 = {};
    const float* pa = Pw + nn * PSTR + 2 * kh;          // A: row m of P
    const float* vb0 = Vt + nn * PSTR + 2 * kh;         // B col n = V^T row n
    const float* vb1 = Vt + (16 + nn) * PSTR + 2 * kh;
    for (int kk = 0; kk < SPAD; kk += 4) {
      v2f a = *(const v2f*)(pa + kk);
      v2f b0 = *(const v2f*)(vb0 + kk);
      v2f b1 = *(const v2f*)(vb1 + kk);
      c0 = wmma4(a, b0, c0);
      c1 = wmma4(a, b1, c1);
    }
#pragma unroll
    for (int r = 0; r < 8; ++r) {
      int q = q0 + r + 8 * kh;
      if (q < L1_SZ) {
        float iv = Sv[w][r + 8 * kh];
        size_t o = (size_t)(b * L1_SZ + q) * HDIM + h * DHEAD;
        Cg[o + nn] = c0[r] * iv;
        Cg[o + 16 + nn] = c1[r] * iv;
      }
    }
  }
}

// ---------------------------------------------------------------------------
// Masked mean over the 200 log positions of interaction_list (GNN collapse).
// ---------------------------------------------------------------------------
__global__ __launch_bounds__(256) void masked_mean_kernel(
    const float* __restrict__ X, const int* __restrict__ mask,
    float* __restrict__ out) {
  int b = blockIdx.x;
  int cnt = 0;
  for (int j = 0; j < LEN; ++j) cnt += (mask[b * L1_SZ + j] != 0);
  float inv = 1.0f / (float)(cnt > 0 ? cnt : 1);
  for (int d = threadIdx.x; d < DLLM; d += blockDim.x) {
    float acc = 0.0f;
    for (int j = 0; j < LEN; ++j)
      if (mask[b * L1_SZ + j]) acc += X[((size_t)b * L1_SZ + j) * DLLM + d];
    out[(size_t)b * DLLM + d] = acc * inv;
  }
}

// ---------------------------------------------------------------------------
// O = LayerNorm(X + Y) over rows of 64. posmode=1: Y is pos_emb[s] (s=r%201).
// One wave per row, 2 elements per lane. In-place (O==X) safe: row-local.
// ---------------------------------------------------------------------------
__global__ __launch_bounds__(256) void add_ln_kernel(
    const float* __restrict__ X, const float* __restrict__ Y,
    const float* __restrict__ g, const float* __restrict__ bta,
    float* __restrict__ O, int rows, int posmode) {
  int lane = threadIdx.x & 31;
  int w = (int)((blockIdx.x * blockDim.x + threadIdx.x) >> 5);
  int nw = (int)((gridDim.x * blockDim.x) >> 5);
  for (int r = w; r < rows; r += nw) {
    const float* xr = X + (size_t)r * HDIM;
    const float* yr = Y + (size_t)(posmode ? (r % L1_SZ) : r) * HDIM;
    float x0 = xr[lane] + yr[lane];
    float x1 = xr[lane + 32] + yr[lane + 32];
    float u = wave_sum(x0 + x1) * (1.0f / 64.0f);
    float d0 = x0 - u, d1 = x1 - u;
    float var = wave_sum(d0 * d0 + d1 * d1) * (1.0f / 64.0f);
    float inv = 1.0f / sqrtf(var + 1e-12f);
    O[(size_t)r * HDIM + lane] = d0 * inv * g[lane] + bta[lane];
    O[(size_t)r * HDIM + lane + 32] = d1 * inv * g[lane + 32] + bta[lane + 32];
  }
}

// ---------------------------------------------------------------------------
// Similarity: sum_i m_i * cos(logs[b,i], base[b]); deterministic two stages.
// ---------------------------------------------------------------------------
__global__ __launch_bounds__(256) void sim_stage1_kernel(
    const float* __restrict__ LOGS, const float* __restrict__ BASE,
    const int* __restrict__ mask, float* __restrict__ part) {
  int b = blockIdx.x;
  int tid = threadIdx.x, w = tid >> 5, lane = tid & 31;
  const float* Bb = BASE + (size_t)b * HDIM;
  float bn2 = 0.0f;
#pragma unroll
  for (int d = 0; d < HDIM; ++d) {
    float t = Bb[d];
    bn2 += t * t;
  }
  float bn = fmaxf(sqrtf(bn2), 1e-8f);
  __shared__ float wpart[8];
  float tot = 0.0f;
  for (int i = w; i < LEN; i += 8) {
    if (mask[b * L1_SZ + i]) {  // uniform across the wave
      const float* Lr = LOGS + ((size_t)b * L1_SZ + i) * HDIM;
      float l0 = Lr[lane], l1 = Lr[lane + 32];
      float dp = wave_sum(l0 * Bb[lane] + l1 * Bb[lane + 32]);
      float n2 = wave_sum(l0 * l0 + l1 * l1);
      if (lane == 0) tot += dp / (fmaxf(sqrtf(n2), 1e-8f) * bn);
    }
  }
  if (lane == 0) wpart[w] = tot;
  __syncthreads();
  if (tid == 0) {
    float s = 0.0f;
    for (int i = 0; i < 8; ++i) s += wpart[i];
    part[b] = s;
  }
}

__global__ __launch_bounds__(256) void sim_stage2_kernel(
    const float* __restrict__ part, float* __restrict__ out) {
  __shared__ float sh[256];
  sh[threadIdx.x] = part[threadIdx.x];
  __syncthreads();
  for (int off = 128; off > 0; off >>= 1) {
    if ((int)threadIdx.x < off) sh[threadIdx.x] += sh[threadIdx.x + off];
    __syncthreads();
  }
  if (threadIdx.x == 0) *out = sh[0] / ((float)B_SZ * (float)LEN);
}

// ---------------------------------------------------------------------------
// Pack all outputs: prec_vec, target_pos, target_neg, combine. One wave/row.
// ---------------------------------------------------------------------------
__global__ __launch_bounds__(256) void pack_kernel(
    const float* __restrict__ EMB, const float* __restrict__ NEG,
    const float* __restrict__ PRC, float* __restrict__ prec_out,
    float* __restrict__ tpos_out, float* __restrict__ tneg_out,
    float* __restrict__ comb_out) {
  int lane = threadIdx.x & 31;
  int w = (int)((blockIdx.x * blockDim.x + threadIdx.x) >> 5);
  int nw = (int)((gridDim.x * blockDim.x) >> 5);
  for (int r = w; r < ROWS; r += nw) {
    int b = r / L1_SZ, s = r % L1_SZ;
    const float* P = PRC + (size_t)r * HDIM;
    const float* E = EMB + (size_t)r * HDIM;
    int sl = (s < LEN) ? s : (LEN - 1);  // last logit duplicated
    const float* Pd = PRC + ((size_t)b * L1_SZ + sl) * HDIM;
    const float* Ed = EMB + ((size_t)b * L1_SZ + sl + 1) * HDIM;
    float dp = wave_sum(Pd[lane] * Ed[lane] + Pd[lane + 32] * Ed[lane + 32]);
    float lg = 1.0f - 1.0f / (1.0f + expf(-dp));
    float* cr = comb_out + (size_t)r * 130;
    cr[lane] = E[lane];
    cr[lane + 32] = E[lane + 32];
    cr[65 + lane] = P[lane];
    cr[65 + lane + 32] = P[lane + 32];
    if (lane == 0) {
      cr[64] = lg;
      cr[129] = lg;
    }
    if (s < LEN) {
      size_t ro = ((size_t)b * LEN + s) * HDIM;
      prec_out[ro + lane] = P[lane];
      prec_out[ro + lane + 32] = P[lane + 32];
      const float* E1 = EMB + ((size_t)b * L1_SZ + s + 1) * HDIM;
      tpos_out[ro + lane] = E1[lane];
      tpos_out[ro + lane + 32] = E1[lane + 32];
      const float* Nr = NEG + (size_t)r * HDIM;
      tneg_out[ro + lane] = Nr[lane];
      tneg_out[ro + lane + 32] = Nr[lane + 32];
    }
  }
}

// ---------------------------------------------------------------------------
extern "C" void kernel_launch(void* const* d_in, const int* in_sizes, int n_in,
                              void* d_out, int out_size, void* d_ws,
                              size_t ws_size, hipStream_t stream) {
  (void)in_sizes; (void)n_in; (void)out_size; (void)ws_size;
  const float* inter = (const float*)d_in[0];
  const float* negl = (const float*)d_in[1];
  const int* maskI = (const int*)d_in[2];
  const float* fc_w = (const float*)d_in[3];
  const float* fc_b = (const float*)d_in[4];
  const float* mlp_w = (const float*)d_in[5];
  const float* mlp_b = (const float*)d_in[6];
  const float* pos_emb = (const float*)d_in[7];
  const float* ln_g = (const float*)d_in[8];
  const float* ln_b = (const float*)d_in[9];
  const float* wq = (const float*)d_in[10];
  const float* wk = (const float*)d_in[11];
  const float* wv = (const float*)d_in[12];
  const float* wo = (const float*)d_in[13];
  const float* bq = (const float*)d_in[14];
  const float* bk = (const float*)d_in[15];
  const float* bv = (const float*)d_in[16];
  const float* bo = (const float*)d_in[17];
  const float* ln1_g = (const float*)d_in[18];
  const float* ln1_b = (const float*)d_in[19];
  const float* ln2_g = (const float*)d_in[20];
  const float* ln2_b = (const float*)d_in[21];
  const float* ffn_w1 = (const float*)d_in[22];
  const float* ffn_b1 = (const float*)d_in[23];
  const float* ffn_w2 = (const float*)d_in[24];
  const float* ffn_b2 = (const float*)d_in[25];
  const float* gnn_w1 = (const float*)d_in[26];
  const float* gnn_b1 = (const float*)d_in[27];
  const float* gnn_w2 = (const float*)d_in[28];
  const float* gnn_b2 = (const float*)d_in[29];

  // Workspace layout (floats)
  float* ws = (float*)d_ws;
  const size_t R = ROWS;
  float* Qb = ws;                  // R*64 (also MLP temp 0, attn-proj/ffn2 out)
  float* Kb = Qb + R * 64;         // R*64 (also MLP temp 1)
  float* Vb = Kb + R * 64;         // R*64
  float* CTX = Vb + R * 64;        // R*64
  float* EMB = CTX + R * 64;       // R*64  embs_all
  float* NEG = EMB + R * 64;       // R*64  neg embeddings
  float* LOGS = NEG + R * 64;      // R*64  logs_full (kept for similarity)
  float* Xb = LOGS + R * 64;       // R*64  encoder state
  float* BIG = Xb + R * 64;        // R*256 ffn hidden
  float* MEANGX = BIG + R * 256;   // 256*768
  float* GT = MEANGX + 256 * 768;  // 256*256
  float* BASE = GT + 256 * 256;    // 256*64
  float* PART = BASE + 256 * 64;   // 256

  const int MR = (int)R;
  auto gemm = [&](const float* A, const float* W, const float* bias, float* C,
                  int M, int N, int K, int act) {
    int mtiles = M / 16;
    int blocks = (mtiles + 7) / 8;
    if (blocks > 2048) blocks = 2048;
    if (blocks < 1) blocks = 1;
    wmma_gemm_kernel<<<blocks, 256, 0, stream>>>(A, W, bias, C, M, N, K, act);
  };
  const int rowBlocks = (MR + 7) / 8;  // wave-per-row kernels

  // ---- _mlp_fc(interaction_list) -> EMB (embs_all)
  gemm(inter, fc_w, fc_b, Qb, MR, 64, 768, 0);
  gemm(Qb, mlp_w, mlp_b, Kb, MR, 64, 64, 1);
  gemm(Kb, mlp_w + 64 * 64, mlp_b + 64, EMB, MR, 64, 64, 1);
  // ---- _mlp_fc(neg_list) -> NEG
  gemm(negl, fc_w, fc_b, Qb, MR, 64, 768, 0);
  gemm(Qb, mlp_w, mlp_b, Kb, MR, 64, 64, 1);
  gemm(Kb, mlp_w + 64 * 64, mlp_b + 64, NEG, MR, 64, 64, 1);

  // ---- collapsed GNN: base[b] = relu(mean_gx@W1+b1)@W2+b2
  masked_mean_kernel<<<B_SZ, 256, 0, stream>>>(inter, maskI, MEANGX);
  gemm(MEANGX, gnn_w1, gnn_b1, GT, B_SZ, 256, 768, 2);
  gemm(GT, gnn_w2, gnn_b2, BASE, B_SZ, 64, 256, 0);

  // ---- logs_full = LN(embs_all + pos_emb)
  add_ln_kernel<<<rowBlocks, 256, 0, stream>>>(EMB, pos_emb, ln_g, ln_b, LOGS,
                                               MR, 1);

  // ---- similarity (independent of encoder)
  sim_stage1_kernel<<<B_SZ, 256, 0, stream>>>(LOGS, BASE, maskI, PART);
  float* out_f = (float*)d_out;
  float* prec_out = out_f;
  float* tpos_out = prec_out + (size_t)B_SZ * LEN * HDIM;
  float* tneg_out = tpos_out + (size_t)B_SZ * LEN * HDIM;
  float* comb_out = tneg_out + (size_t)B_SZ * LEN * HDIM;
  float* sim_out = comb_out + (size_t)B_SZ * L1_SZ * 130;
  sim_stage2_kernel<<<1, 256, 0, stream>>>(PART, sim_out);

  // ---- transformer encoder on S=201 (prec_vec is its causal prefix)
  hipMemcpyAsync(Xb, LOGS, R * 64 * sizeof(float), hipMemcpyDeviceToDevice,
                 stream);
  for (int i = 0; i < 2; ++i) {
    gemm(Xb, wq + i * 4096, bq + i * 64, Qb, MR, 64, 64, 0);
    gemm(Xb, wk + i * 4096, bk + i * 64, Kb, MR, 64, 64, 0);
    gemm(Xb, wv + i * 4096, bv + i * 64, Vb, MR, 64, 64, 0);
    attention_wmma_kernel<<<B_SZ * NHEAD, 128, 0, stream>>>(Qb, Kb, Vb, maskI,
                                                            CTX);
    gemm(CTX, wo + i * 4096, bo + i * 64, Qb, MR, 64, 64, 0);
    add_ln_kernel<<<rowBlocks, 256, 0, stream>>>(Xb, Qb, ln1_g + i * 64,
                                                 ln1_b + i * 64, Xb, MR, 0);
    gemm(Xb, ffn_w1 + i * 64 * 256, ffn_b1 + i * 256, BIG, MR, 256, 64, 1);
    gemm(BIG, ffn_w2 + i * 256 * 64, ffn_b2 + i * 64, Qb, MR, 64, 256, 0);
    add_ln_kernel<<<rowBlocks, 256, 0, stream>>>(Xb, Qb, ln2_g + i * 64,
                                                 ln2_b + i * 64, Xb, MR, 0);
  }

  // ---- pack prec_vec / target_pos / target_neg / combine
  pack_kernel<<<rowBlocks, 256, 0, stream>>>(EMB, NEG, Xb, prec_out, tpos_out,
                                             tneg_out, comb_out);
}